// Dilate_5592047419617
// MI455X (gfx1250) — compile-verified
//
#include <hip/hip_runtime.h>

// Grayscale 3x3 dilation (kornia Dilate semantics) over (8,3,1024,1024) f32.
//   out[y,x] = max_i ( (se[i]-1 >= 0) ? padded[y + i/3, x + i%3] + (se[i]-1)
//                                     : (se[i]-1) )
// Zero padding of 1 on each side. Memory-bound (~200 MB @ 23.3 TB/s => ~9 us
// floor): stage rows in LDS via the CDNA5 async global->LDS path (ASYNCcnt)
// so HBM reads are (nearly) once-only; TH=8 tile => 10/8 = 1.25x halo
// amplification. NT stores keep L2 free for input halo reuse.

#define B_    8
#define C_    3
#define H_    1024
#define W_    1024
#define TH_   8              // output rows per block
#define SROWS 10             // staged rows = TH_ + 2 (halo)
#define SROWF 1032           // floats per LDS row: 4128 B, multiple of 16 B
#define DCOL  4              // float index where column 0 lives (byte 16)

typedef float v4f __attribute__((ext_vector_type(4)));   // native vector type

__global__ __launch_bounds__(256) void dilate3x3_kernel(
    const float* __restrict__ in, const float* __restrict__ se,
    float* __restrict__ out)
{
    __shared__ __align__(16) float smem[SROWS * SROWF];   // 41280 B

    const int tid  = threadIdx.x;                 // 0..255
    const int tilesPerImg = H_ / TH_;             // 128
    const int tile = blockIdx.x % tilesPerImg;    // uniform
    const int img  = blockIdx.x / tilesPerImg;    // 0..23, uniform
    const int y0   = tile * TH_;

    const float* imgBase = in + (size_t)img * (size_t)(H_ * W_);

    // LDS byte offset of smem base: flat LDS aperture address truncated to
    // 32 bits is the LDS offset (ISA 10.2 aperture rules).
    const unsigned ldsBase = (unsigned)(unsigned long long)(&smem[0]);

    // ---- Stage rows y0-1 .. y0+TH into LDS (async, 16 B per lane per row) ----
    #pragma unroll
    for (int j = 0; j < SROWS; ++j) {
        const int yin = y0 - 1 + j;               // uniform
        const unsigned ldsOff =
            ldsBase + (unsigned)((j * SROWF + DCOL) * 4) + (unsigned)(tid * 16);
        if (yin >= 0 && yin < H_) {               // uniform branch: EXEC all-1s
            const float* rowPtr = imgBase + (size_t)yin * W_;  // uniform -> SGPR
            const unsigned voff = (unsigned)(tid * 16);        // per-lane offset
            asm volatile("global_load_async_to_lds_b128 %0, %1, %2"
                         :: "v"(ldsOff), "v"(voff), "s"(rowPtr)
                         : "memory");
        } else {
            v4f z = (v4f)(0.0f);
            *(v4f*)&smem[j * SROWF + DCOL + tid * 4] = z;      // zero row pad
        }
    }
    // Horizontal zero padding (columns -1 and W) for each staged row.
    if (tid < SROWS) {
        smem[tid * SROWF + (DCOL - 1)] = 0.f;
        smem[tid * SROWF + DCOL + W_]  = 0.f;
    }

    asm volatile("s_wait_asynccnt 0" ::: "memory");  // drain this wave's async
    __syncthreads();                                  // publish LDS to all waves

    // ---- SE offsets (uniform scalar loads) ----
    float m[9];
    #pragma unroll
    for (int i = 0; i < 9; ++i) m[i] = se[i] - 1.0f;

    // ---- Sliding 3-row x 6-col register window over the staged tile ----
    const int c0 = tid * 4;                      // this thread's 4 columns
    float w[3][6];                               // w[r%3] holds lds row r
    #pragma unroll
    for (int j = 0; j < 2; ++j) {
        const float* r = &smem[j * SROWF + DCOL + c0 - 1];
        #pragma unroll
        for (int q = 0; q < 6; ++q) w[j][q] = r[q];
    }

    float* outImg = out + (size_t)img * (size_t)(H_ * W_);
    #pragma unroll
    for (int rr = 0; rr < TH_; ++rr) {
        // bring in lds row rr+2
        {
            const float* r = &smem[(rr + 2) * SROWF + DCOL + c0 - 1];
            #pragma unroll
            for (int q = 0; q < 6; ++q) w[(rr + 2) % 3][q] = r[q];
        }
        float res[4];
        #pragma unroll
        for (int kk = 0; kk < 4; ++kk) {
            float best;
            #pragma unroll
            for (int i = 0; i < 9; ++i) {
                const int dy = i / 3, dx = i % 3;   // row-major map, matching the header formula
                const float v = w[(rr + dy) % 3][kk + dx];
                const float contrib = (m[i] >= 0.0f) ? (v + m[i]) : m[i];
                best = (i == 0) ? contrib : fmaxf(best, contrib);
            }
            res[kk] = best;
        }
        v4f o;
        o.x = res[0]; o.y = res[1]; o.z = res[2]; o.w = res[3];
        // outputs are never re-read: nontemporal store keeps L2 for the input
        __builtin_nontemporal_store(o, (v4f*)&outImg[(size_t)(y0 + rr) * W_ + c0]);
    }
}

extern "C" void kernel_launch(void* const* d_in, const int* in_sizes, int n_in,
                              void* d_out, int out_size, void* d_ws, size_t ws_size,
                              hipStream_t stream) {
    (void)in_sizes; (void)n_in; (void)d_ws; (void)ws_size; (void)out_size;
    const float* in = (const float*)d_in[0];   // (8,3,1024,1024) f32
    const float* se = (const float*)d_in[1];   // (3,3) f32
    float* out = (float*)d_out;                // (8,3,1024,1024) f32

    const int blocks = B_ * C_ * (H_ / TH_);   // 24 * 128 = 3072
    dilate3x3_kernel<<<dim3(blocks), dim3(256), 0, stream>>>(in, se, out);
}